// QuadRasterizer_48430051230190
// MI455X (gfx1250) — compile-verified
//
#include <hip/hip_runtime.h>

// Soft rasterizer blend (pytorch3d softmax_rgb_blend style) for MI455X/gfx1250.
// One thread per pixel. Fragment metadata (pix_to_face / zbuf / dists) is
// staged block-wide into LDS via CDNA5 async global->LDS DMA (ASYNCcnt path).
// Vertex-feature gathers are skipped for fragments whose softmax weight
// underflows to exactly 0 (gamma = 1e-5 => ~1 of 8 fragments contributes),
// which is bit-identical to the reference and cuts gather traffic ~8x.

#define SIGMA_F 1e-5f
#define GAMMA_F 1e-5f
#define ZNEAR_F 0.1f
#define ZFAR_F  10.0f
#define EPS_F   1e-10f

#define HF 512
#define WF 512
#define KF 8
#define DF 16
#define HWPIX (HF * WF)

#define BLK 256

// LDS layout (padded strides to dodge power-of-2 bank conflicts)
#define P2F_STRIDE 80          // 64B payload + 16B pad
#define Z_STRIDE   48          // 32B payload + 16B pad
#define LDS_P2F    0
#define LDS_Z      (BLK * P2F_STRIDE)          // 20480
#define LDS_DI     (LDS_Z + BLK * Z_STRIDE)    // 32768
#define LDS_BYTES  (LDS_DI + BLK * Z_STRIDE)   // 45056

#if defined(__has_builtin)
#  if __has_builtin(__builtin_amdgcn_global_load_async_to_lds_b128)
#    define USE_ASYNC_BUILTIN 1
#  endif
#  if __has_builtin(__builtin_amdgcn_s_wait_asynccnt)
#    define USE_WAITASYNC_BUILTIN 1
#  endif
#endif

typedef int v4i_t __attribute__((ext_vector_type(4)));
typedef __attribute__((address_space(1))) v4i_t* g_v4i_p;   // global (device)
typedef __attribute__((address_space(3))) v4i_t* l_v4i_p;   // LDS (shared)
typedef __attribute__((address_space(3))) char*  lds_cptr_t;

__device__ __forceinline__ void async_g2lds_b128(const void* g, void* l) {
#if defined(__HIP_DEVICE_COMPILE__)
#  ifdef USE_ASYNC_BUILTIN
  __builtin_amdgcn_global_load_async_to_lds_b128(
      (g_v4i_p)(char*)g,   // AS1 v4i* : global source (16B / lane)
      (l_v4i_p)(char*)l,   // AS3 v4i* : LDS destination
      0, 0);               // imm offset, cpol
#  else
  unsigned loff =
      (unsigned)(unsigned long long)(lds_cptr_t)(char*)l;    // LDS byte offset
  unsigned long long ga = (unsigned long long)g;             // 64-bit global addr
  asm volatile("global_load_async_to_lds_b128 %0, %1, off"
               :: "v"(loff), "v"(ga) : "memory");
#  endif
#else
  (void)g; (void)l;
#endif
}

__device__ __forceinline__ void wait_async_zero() {
#if defined(__HIP_DEVICE_COMPILE__)
#  ifdef USE_WAITASYNC_BUILTIN
  __builtin_amdgcn_s_wait_asynccnt(0);
#  else
  asm volatile("s_wait_asynccnt 0" ::: "memory");
#  endif
#endif
}

__global__ __launch_bounds__(BLK) void softras_blend_kernel(
    const float* __restrict__ verts_feats,   // [V,16]
    const float* __restrict__ bary,          // [HW,8,3]
    const float* __restrict__ zbuf,          // [HW,8]
    const float* __restrict__ dists,         // [HW,8]
    const long long* __restrict__ faces,     // [F,3]
    const long long* __restrict__ p2f,       // [HW,8]
    float* __restrict__ out)                 // [17,HW]
{
  __shared__ __align__(16) char smem[LDS_BYTES];

  const int tid = threadIdx.x;
  const int pb  = blockIdx.x * BLK;         // first pixel of this block
  const int p   = pb + tid;                 // this thread's pixel

  // ---- Stage fragment metadata block-wide into LDS (async DMA, b128/lane) --
  {
    const char* gp = (const char*)p2f + (size_t)pb * 64;   // 64B per pixel
#pragma unroll
    for (int it = 0; it < 4; ++it) {
      int c = tid + it * BLK;               // 16B chunk index, 0..1023
      int pix = c >> 2, j = c & 3;
      async_g2lds_b128(gp + (size_t)c * 16,
                       smem + LDS_P2F + pix * P2F_STRIDE + j * 16);
    }
    const char* gz = (const char*)zbuf  + (size_t)pb * 32; // 32B per pixel
    const char* gd = (const char*)dists + (size_t)pb * 32;
#pragma unroll
    for (int it = 0; it < 2; ++it) {
      int c = tid + it * BLK;               // 0..511
      int pix = c >> 1, j = c & 1;
      async_g2lds_b128(gz + (size_t)c * 16, smem + LDS_Z  + pix * Z_STRIDE + j * 16);
      async_g2lds_b128(gd + (size_t)c * 16, smem + LDS_DI + pix * Z_STRIDE + j * 16);
    }
  }
  wait_async_zero();
  __syncthreads();

  const long long* lp2f = (const long long*)(smem + LDS_P2F + tid * P2F_STRIDE);
  const float*     lz   = (const float*)(smem + LDS_Z  + tid * Z_STRIDE);
  const float*     ld   = (const float*)(smem + LDS_DI + tid * Z_STRIDE);

  // ---- Pass 1: coverage prob, inverse depth, depth max ----------------------
  float prob[KF], zinv[KF];
  float zmax = EPS_F;
#pragma unroll
  for (int k = 0; k < KF; ++k) {
    bool valid = lp2f[k] >= 0;
    float pr = 1.0f / (1.0f + expf(ld[k] / SIGMA_F));   // sigmoid(-d/sigma)
    prob[k] = valid ? pr : 0.0f;
    float zi = valid ? (ZFAR_F - lz[k]) / (ZFAR_F - ZNEAR_F) : 0.0f;
    zinv[k] = zi;
    zmax = fmaxf(zmax, zi);
  }

  // ---- Pass 2: softmax weights, denominator, alpha --------------------------
  float wnum[KF];
  float wsum = 0.0f, om = 1.0f;
#pragma unroll
  for (int k = 0; k < KF; ++k) {
    float wv = prob[k] * expf((zinv[k] - zmax) / GAMMA_F);
    wnum[k] = wv;
    wsum += wv;
    om *= (1.0f - prob[k]);
  }
  const float denom = wsum + expf((EPS_F - zmax) / GAMMA_F);

  // ---- Pass 3: gather + weighted barycentric interpolation ------------------
  // Skip fragments whose weight is exactly 0 (their contribution is exactly 0
  // in the reference as well) -> ~8x fewer vertex gathers.
  float acc[DF];
#pragma unroll
  for (int d = 0; d < DF; ++d) acc[d] = 0.0f;

  const float* bb = bary + (size_t)p * (KF * 3);
#pragma unroll
  for (int k = 0; k < KF; ++k) {
    float wv = wnum[k];
    if (wv > 0.0f) {
      long long f = lp2f[k];                       // valid (prob>0 => f>=0)
      const long long* fvi = faces + f * 3;
      long long i0 = fvi[0], i1 = fvi[1], i2 = fvi[2];
      float w0 = wv * bb[k * 3 + 0];
      float w1 = wv * bb[k * 3 + 1];
      float w2 = wv * bb[k * 3 + 2];
      const float4* v0 = (const float4*)(verts_feats + (size_t)i0 * DF);
      const float4* v1 = (const float4*)(verts_feats + (size_t)i1 * DF);
      const float4* v2 = (const float4*)(verts_feats + (size_t)i2 * DF);
#pragma unroll
      for (int q = 0; q < 4; ++q) {               // 4 x b128 per vertex row
        float4 t0 = v0[q], t1 = v1[q], t2 = v2[q];
        acc[4 * q + 0] += w0 * t0.x + w1 * t1.x + w2 * t2.x;
        acc[4 * q + 1] += w0 * t0.y + w1 * t1.y + w2 * t2.y;
        acc[4 * q + 2] += w0 * t0.z + w1 * t1.z + w2 * t2.z;
        acc[4 * q + 3] += w0 * t0.w + w1 * t1.w + w2 * t2.w;
      }
    }
  }

  // ---- Stores: channel-major [17, H*W], coalesced across lanes --------------
  const float inv = 1.0f / denom;
#pragma unroll
  for (int d = 0; d < DF; ++d)
    out[(size_t)d * HWPIX + p] = acc[d] * inv;
  out[(size_t)DF * HWPIX + p] = 1.0f - om;
}

extern "C" void kernel_launch(void* const* d_in, const int* in_sizes, int n_in,
                              void* d_out, int out_size, void* d_ws, size_t ws_size,
                              hipStream_t stream) {
  (void)in_sizes; (void)n_in; (void)out_size; (void)d_ws; (void)ws_size;
  const float*     verts_feats = (const float*)d_in[0];
  const float*     bary        = (const float*)d_in[1];
  const float*     zbuf        = (const float*)d_in[2];
  const float*     dists       = (const float*)d_in[3];
  const long long* faces       = (const long long*)d_in[4];
  const long long* p2f         = (const long long*)d_in[5];
  float*           out         = (float*)d_out;

  dim3 grid(HWPIX / BLK);   // 262144 / 256 = 1024 blocks, exact cover
  dim3 block(BLK);
  hipLaunchKernelGGL(softras_blend_kernel, grid, block, 0, stream,
                     verts_feats, bary, zbuf, dists, faces, p2f, out);
}